// GCN_49838800503557
// MI455X (gfx1250) — compile-verified
//
#include <hip/hip_runtime.h>

typedef __attribute__((ext_vector_type(2))) float v2f;
typedef __attribute__((ext_vector_type(8))) float v8f;

#define N_NODES 100000
#define N_EDGES 1600000
#define F 128
#define N_OUT 4

// ---------------------------------------------------------------------------
// One-time weight transpose: Wt[mtx][c][k] = W[mtx][k][c] so that the WMMA
// B-fragment (two consecutive K values at a fixed output column) is a single
// contiguous 8-byte load. Also transposes W_out (128x4 -> 4x128).
// ---------------------------------------------------------------------------
__global__ void transpose_weights(const float* __restrict__ W_in,
                                  const float* __restrict__ W_hid,
                                  const float* __restrict__ W_out,
                                  float* __restrict__ Wt) {
    int i = blockIdx.x * blockDim.x + threadIdx.x;
    if (i < 6 * F * F) {
        int mtx = i / (F * F);
        int rem = i % (F * F);
        int c = rem / F;        // output column (row of Wt)
        int k = rem % F;        // K index (col of Wt)
        const float* Wsrc = (mtx == 0) ? W_in : (W_hid + (size_t)(mtx - 1) * F * F);
        Wt[i] = Wsrc[(size_t)k * F + c];
    } else if (i < 6 * F * F + N_OUT * F) {
        int j = i - 6 * F * F;
        int c = j / F;          // 0..3
        int k = j % F;
        Wt[i] = W_out[(size_t)k * N_OUT + c];
    }
}

// ---------------------------------------------------------------------------
// Dense GEMM: out[N x 128] = relu?(in)[N x 128] @ W[128 x 128], f32 WMMA
// 16x16x4. Block = 256 threads = 8 waves; one 16-row tile per block, one
// 16-col strip per wave. A tile staged in LDS with ReLU fused into the stage
// (the previous layer's activation is applied here, saving a full 102MB/layer
// elementwise pass). Wt is column-major so B fragments are b64 loads.
// 100000 rows = 6250 blocks exactly -> no guards, EXEC all-ones for WMMA.
// ---------------------------------------------------------------------------
__global__ __launch_bounds__(256)
void gemm128_wmma(const float* __restrict__ in, const float* __restrict__ Wt,
                  float* __restrict__ out, int doRelu) {
    __shared__ float lds[16 * F];
    const int rowBase = blockIdx.x * 16;
    {
        // 16*128 = 2048 floats = 512 float4; 256 threads x 2, ReLU fused
        const float4* g = (const float4*)(in + (size_t)rowBase * F);
        float4* l4 = (float4*)lds;
        float4 v0 = g[threadIdx.x];
        float4 v1 = g[threadIdx.x + 256];
        if (doRelu) {
            v0.x = fmaxf(v0.x, 0.0f); v0.y = fmaxf(v0.y, 0.0f);
            v0.z = fmaxf(v0.z, 0.0f); v0.w = fmaxf(v0.w, 0.0f);
            v1.x = fmaxf(v1.x, 0.0f); v1.y = fmaxf(v1.y, 0.0f);
            v1.z = fmaxf(v1.z, 0.0f); v1.w = fmaxf(v1.w, 0.0f);
        }
        l4[threadIdx.x]       = v0;
        l4[threadIdx.x + 256] = v1;
    }
    __syncthreads();

    const int wave = threadIdx.x >> 5;      // 0..7 -> column tile
    const int lane = threadIdx.x & 31;
    const int m    = lane & 15;
    const int half = lane >> 4;             // 0 or 1
    const int koff = half * 2;              // K sub-offset per ISA layout
    const int col  = wave * 16 + m;
    const float* __restrict__ wcol = Wt + (size_t)col * F;   // column of W

    v8f c = {};
#pragma unroll
    for (int k0 = 0; k0 < F; k0 += 4) {
        const int kb = k0 + koff;
        // A 16x4 f32: lanes 0-15 hold K=k0,k0+1; lanes 16-31 hold K=k0+2,k0+3
        v2f a = *(const v2f*)(&lds[m * F + kb]);
        // B 4x16 f32: contiguous K pair for this column -> single b64 load
        v2f b = *(const v2f*)(&wcol[kb]);
        c = __builtin_amdgcn_wmma_f32_16x16x4_f32(false, a, false, b,
                                                  (short)0, c, false, false);
    }
    // D 16x16 f32: lanes 0-15 -> M=r, lanes 16-31 -> M=8+r
#pragma unroll
    for (int r = 0; r < 8; ++r) {
        out[(size_t)(rowBase + r + half * 8) * F + col] = c[r];
    }
}

// ---------------------------------------------------------------------------
// Output GEMM: out[N x 4] = relu(in)[N x 128] @ Wo[128 x 4], N padded to 16
// with zero columns. EXEC kept uniform through the WMMA (clamped loads +
// select); only cols 0..3 stored. WtO is 4x128 (transposed) -> b64 B loads.
// One wave per 16-row tile, 6250 blocks exact.
// ---------------------------------------------------------------------------
__global__ __launch_bounds__(32)
void gemm_out_wmma(const float* __restrict__ in, const float* __restrict__ WtO,
                   float* __restrict__ out) {
    const int rowBase = blockIdx.x * 16;
    const int lane = threadIdx.x & 31;
    const int m    = lane & 15;
    const int half = lane >> 4;
    const int koff = half * 2;
    const int colc = (m < N_OUT) ? m : (N_OUT - 1);  // clamp: keep EXEC uniform
    const float sel = (m < N_OUT) ? 1.0f : 0.0f;
    const float* __restrict__ arow = in + (size_t)(rowBase + m) * F;
    const float* __restrict__ wcol = WtO + (size_t)colc * F;

    v8f c = {};
#pragma unroll
    for (int k0 = 0; k0 < F; k0 += 4) {
        const int kb = k0 + koff;
        v2f a = *(const v2f*)(&arow[kb]);
        a.x = fmaxf(a.x, 0.0f);                      // fused ReLU of last hidden
        a.y = fmaxf(a.y, 0.0f);
        v2f b = *(const v2f*)(&wcol[kb]);
        b.x *= sel;
        b.y *= sel;
        c = __builtin_amdgcn_wmma_f32_16x16x4_f32(false, a, false, b,
                                                  (short)0, c, false, false);
    }
    if (m < N_OUT) {
#pragma unroll
        for (int r = 0; r < 8; ++r) {
            out[(size_t)(rowBase + r + half * 8) * N_OUT + m] = c[r];
        }
    }
}

// ---------------------------------------------------------------------------
// Degree / normalization prep
// ---------------------------------------------------------------------------
__global__ void zero_f32(float* __restrict__ p, int n) {
    int i = blockIdx.x * blockDim.x + threadIdx.x;
    if (i < n) p[i] = 0.0f;
}

__global__ void count_deg(const int* __restrict__ dst, float* __restrict__ deg, int nE) {
    int i = blockIdx.x * blockDim.x + threadIdx.x;
    if (i < nE) atomicAdd(&deg[dst[i]], 1.0f);
}

__global__ void make_dinv(const float* __restrict__ deg,
                          float* __restrict__ dinv2, float* __restrict__ dinv1, int n) {
    int i = blockIdx.x * blockDim.x + threadIdx.x;
    if (i < n) {
        float d = deg[i];
        dinv2[i] = rsqrtf(d + 2.0f);
        dinv1[i] = rsqrtf(d + 1.0f);
    }
}

// ---------------------------------------------------------------------------
// agg init: agg = fill*dinv^2*xw + bias   (self-loop + bias folded in)
// ---------------------------------------------------------------------------
__global__ void init_agg128(const float* __restrict__ xw, const float* __restrict__ dinv,
                            const float* __restrict__ bias, float fill,
                            float* __restrict__ agg) {
    int i = blockIdx.x * blockDim.x + threadIdx.x;
    if (i >= N_NODES * 32) return;
    int node = i >> 5;
    int q    = i & 31;
    float dv = dinv[node];
    float s  = fill * dv * dv;
    float4 v  = ((const float4*)xw)[i];
    float4 bb = ((const float4*)bias)[q];
    float4 o;
    o.x = s * v.x + bb.x;
    o.y = s * v.y + bb.y;
    o.z = s * v.z + bb.z;
    o.w = s * v.w + bb.w;
    ((float4*)agg)[i] = o;
}

// ---------------------------------------------------------------------------
// Edge scatter: one edge per wave, 32 lanes x float4 = 128 features.
// xw rows are L2-resident (51MB < 192MB L2); atomics resolve in L2.
// ---------------------------------------------------------------------------
__global__ __launch_bounds__(256)
void scatter128(const int* __restrict__ src, const int* __restrict__ dst,
                const float* __restrict__ dinv, const float* __restrict__ xw,
                float* __restrict__ agg, int nE) {
    int e = (blockIdx.x * blockDim.x + threadIdx.x) >> 5;
    if (e >= nE) return;
    int lane = threadIdx.x & 31;
    int s = src[e];
    int d = dst[e];
    float coef = dinv[s] * dinv[d];
    float4 v = ((const float4*)(xw + (size_t)s * F))[lane];
    float* ad = agg + (size_t)d * F + lane * 4;
    atomicAdd(ad + 0, coef * v.x);
    atomicAdd(ad + 1, coef * v.y);
    atomicAdd(ad + 2, coef * v.z);
    atomicAdd(ad + 3, coef * v.w);
}

// Final layer (fill = 1): init d_out with self-loop + bias, then edge atomics.
__global__ void init_out(const float* __restrict__ xwo, const float* __restrict__ dinv,
                         const float* __restrict__ bo, float* __restrict__ out) {
    int i = blockIdx.x * blockDim.x + threadIdx.x;
    if (i >= N_NODES * N_OUT) return;
    int node = i >> 2;
    int j    = i & 3;
    float dv = dinv[node];
    out[i] = dv * dv * xwo[i] + bo[j];
}

__global__ void scatter4(const int* __restrict__ src, const int* __restrict__ dst,
                         const float* __restrict__ dinv, const float* __restrict__ xwo,
                         float* __restrict__ out, int nE) {
    int e = blockIdx.x * blockDim.x + threadIdx.x;
    if (e >= nE) return;
    int s = src[e];
    int d = dst[e];
    float coef = dinv[s] * dinv[d];
    float4 v = ((const float4*)xwo)[s];
    float* od = out + (size_t)d * N_OUT;
    atomicAdd(od + 0, coef * v.x);
    atomicAdd(od + 1, coef * v.y);
    atomicAdd(od + 2, coef * v.z);
    atomicAdd(od + 3, coef * v.w);
}

// ---------------------------------------------------------------------------
extern "C" void kernel_launch(void* const* d_in, const int* in_sizes, int n_in,
                              void* d_out, int out_size, void* d_ws, size_t ws_size,
                              hipStream_t stream) {
    (void)in_sizes; (void)n_in; (void)out_size; (void)ws_size;

    const float* x     = (const float*)d_in[0];
    const int*   ei    = (const int*)d_in[1];           // [2, N_EDGES]
    const float* W_in  = (const float*)d_in[2];
    const float* b_in  = (const float*)d_in[3];
    const float* W_hid = (const float*)d_in[4];         // [5,128,128]
    const float* b_hid = (const float*)d_in[5];         // [5,128]
    const float* W_out = (const float*)d_in[6];         // [128,4]
    const float* b_out = (const float*)d_in[7];
    float* out = (float*)d_out;

    const int* src = ei;
    const int* dst = ei + N_EDGES;

    const size_t nodesF = (size_t)N_NODES * F;          // 12.8M floats
    float* B     = (float*)d_ws;                        // xw     (51.2 MB)
    float* C     = B + nodesF;                          // agg / h (51.2 MB)
    float* deg   = C + nodesF;                          // 100k
    float* dinv2 = deg + N_NODES;                       // rsqrt(deg+2)
    float* dinv1 = dinv2 + N_NODES;                     // rsqrt(deg+1)
    float* xwo   = dinv1 + N_NODES;                     // 100k x 4
    float* Wt    = xwo + (size_t)N_NODES * N_OUT;       // 6*128*128 + 4*128

    const int nodeBlk  = (N_NODES + 255) / 256;
    const int edgeBlk  = (N_EDGES + 255) / 256;
    const int featBlk  = (N_NODES * 32 + 255) / 256;    // float4 granularity
    const int gemmBlk  = N_NODES / 16;                  // 6250 exact
    const int scatBlk  = (N_EDGES + 7) / 8;             // 8 waves/block, 1 edge/wave
    const int wtBlk    = (6 * F * F + N_OUT * F + 255) / 256;

    // one-time prep: degree tables + transposed weights
    transpose_weights<<<wtBlk, 256, 0, stream>>>(W_in, W_hid, W_out, Wt);
    zero_f32<<<nodeBlk, 256, 0, stream>>>(deg, N_NODES);
    count_deg<<<edgeBlk, 256, 0, stream>>>(dst, deg, N_EDGES);
    make_dinv<<<nodeBlk, 256, 0, stream>>>(deg, dinv2, dinv1, N_NODES);

    // 6 GCN layers (fill = 2.0); ReLU of layer i is fused into layer i+1's
    // GEMM A-stage, so C (pre-ReLU agg) is the inter-layer activation buffer.
    for (int i = 0; i < 6; ++i) {
        const float* hin = (i == 0) ? x : C;
        const float* Wti = Wt + (size_t)i * F * F;
        const float* bb  = (i == 0) ? b_in : (b_hid + (size_t)(i - 1) * F);
        gemm128_wmma<<<gemmBlk, 256, 0, stream>>>(hin, Wti, B, i > 0 ? 1 : 0);
        init_agg128<<<featBlk, 256, 0, stream>>>(B, dinv2, bb, 2.0f, C);
        scatter128<<<scatBlk, 256, 0, stream>>>(src, dst, dinv2, B, C, N_EDGES);
    }

    // output layer (fill = 1.0, no ReLU on result; ReLU of last hidden fused
    // into gemm_out's A loads), accumulate straight into d_out
    gemm_out_wmma<<<gemmBlk, 32, 0, stream>>>(C, Wt + 6 * F * F, xwo);
    init_out<<<(N_NODES * N_OUT + 255) / 256, 256, 0, stream>>>(xwo, dinv1, b_out, out);
    scatter4<<<edgeBlk, 256, 0, stream>>>(src, dst, dinv1, xwo, out, N_EDGES);
}